// TopoGCN_66589172957710
// MI455X (gfx1250) — compile-verified
//
#include <hip/hip_runtime.h>

// ---------------------------------------------------------------------------
// 2-layer GCN (PyG-style, symmetric norm, self loops) + global mean pool.
//
//   deg[i]  = 1 + #edges with dst==i           (self loop included)
//   dinv[i] = rsqrt(deg[i])                    (deg >= 1 always)
//   layer:   hs = (X W) * dinv   (per node)
//            acc[i] = hs[i]                    (self-loop contribution)
//            acc[d] += hs[s]   for each edge   (scatter-add, f32 atomics in L2)
//            out[i] = acc[i]*dinv[i] + b
//   Layer-2 node transform x2[N,16] @ W2[16,2] on V_WMMA_F32_16X16X4_F32
//   (16-node tile per wave, K=16 as 4 chained K=4 WMMAs, W2 padded to 4x16).
// ---------------------------------------------------------------------------

typedef float v2f __attribute__((ext_vector_type(2)));
typedef float v8f __attribute__((ext_vector_type(8)));

constexpr int kNodes  = 100000;
constexpr int kEdges  = 6400000;
constexpr int kGraphs = 64;

// --- init: deg counters = 1 (self loop), pool sums/counts = 0 --------------
__global__ void gcn_init_kernel(unsigned* __restrict__ degu,
                                float* __restrict__ sums,
                                float* __restrict__ cnt, int n) {
  int i = blockIdx.x * blockDim.x + threadIdx.x;
  if (i < n) degu[i] = 1u;
  if (i < 2 * kGraphs) sums[i] = 0.0f;
  if (i < kGraphs) cnt[i] = 0.0f;
}

// --- degree accumulation over real edges -----------------------------------
__global__ void gcn_deg_kernel(const int* __restrict__ dst,
                               unsigned* __restrict__ degu, int e) {
  int i = blockIdx.x * blockDim.x + threadIdx.x;
  if (i < e) atomicAdd(&degu[dst[i]], 1u);
}

// --- layer-1 node transform: hs1 = (x*dinv) * W1 ; acc1 = hs1 --------------
__global__ void gcn_node_l1_kernel(const float* __restrict__ x,
                                   const unsigned* __restrict__ degu,
                                   const float* __restrict__ W1,
                                   float* __restrict__ dinv,
                                   float* __restrict__ hs1,
                                   float* __restrict__ acc1, int n) {
  int i = blockIdx.x * blockDim.x + threadIdx.x;
  if (i >= n) return;
  float di = rsqrtf((float)degu[i]);   // deg >= 1 always (self loop)
  dinv[i] = di;
  float xs = x[i] * di;
#pragma unroll
  for (int q = 0; q < 4; ++q) {
    float4 w = ((const float4*)W1)[q];
    float4 r;
    r.x = xs * w.x; r.y = xs * w.y; r.z = xs * w.z; r.w = xs * w.w;
    ((float4*)(hs1 + (size_t)i * 16))[q]  = r;
    ((float4*)(acc1 + (size_t)i * 16))[q] = r;
  }
}

// --- layer-1 edge scatter: acc1[dst] += hs1[src] (16 feats) ----------------
__global__ void gcn_scatter_l1_kernel(const int* __restrict__ src,
                                      const int* __restrict__ dst,
                                      const float* __restrict__ hs1,
                                      float* __restrict__ acc1, int e) {
  int i = blockIdx.x * blockDim.x + threadIdx.x;
  if (i >= e) return;
  int s = src[i], d = dst[i];
  const float4* hp = (const float4*)(hs1 + (size_t)s * 16);
  float* ap = acc1 + (size_t)d * 16;
#pragma unroll
  for (int q = 0; q < 4; ++q) {
    float4 m = hp[q];
    atomicAdd(ap + 4 * q + 0, m.x);
    atomicAdd(ap + 4 * q + 1, m.y);
    atomicAdd(ap + 4 * q + 2, m.z);
    atomicAdd(ap + 4 * q + 3, m.w);
  }
}

// --- layer-2 node transform via WMMA f32 16x16x4 ---------------------------
// One wave = one 16-node tile. x2[i][j] = acc1[i][j]*dinv[i] + b1[j].
// A frag (16x4, k-chunk k): lanes 0-15  -> {x2[row][4k+0], x2[row][4k+1]}
//                           lanes 16-31 -> {x2[row][4k+2], x2[row][4k+3]}
// B frag (4x16): col = lane&15; VGPR0 = rows {4k+0 | 4k+2}, VGPR1 = +1.
// D: vgpr r, lanes 0-15 -> D[r][lane]; lanes 16-31 -> D[r+8][lane-16].
__global__ __launch_bounds__(256)
void gcn_l2_wmma_kernel(const float* __restrict__ acc1,
                        const float* __restrict__ dinv,
                        const float* __restrict__ b1,
                        const float* __restrict__ W2,
                        float* __restrict__ hs2,
                        float* __restrict__ acc2, int nTiles) {
  const int lane = threadIdx.x & 31;
  const int wave = threadIdx.x >> 5;
  const int tile = blockIdx.x * 8 + wave;
  if (tile >= nTiles) return;              // wave-uniform: EXEC all-ones below
  const int nb   = tile * 16;
  const int row  = lane & 15;              // node row for A / column for B,D
  const int node = nb + row;
  const float di = dinv[node];
  const int koff = (lane >> 4) << 1;       // 0 for lanes 0-15, 2 for 16-31
  const float* xrow = acc1 + (size_t)node * 16;

  // Branchless B fragments: clamp the column index so every lane issues a
  // valid load (no exec-mask predication), then zero via multiply-select.
  const float cmask = (row < 2) ? 1.0f : 0.0f;
  const int   cidx  = (row < 2) ? row : 0;
  v2f bf[4];
#pragma unroll
  for (int k = 0; k < 4; ++k) {
    const int kb = 4 * k + koff;
    bf[k].x = W2[(kb + 0) * 2 + cidx] * cmask;
    bf[k].y = W2[(kb + 1) * 2 + cidx] * cmask;
  }

  // A fragments: x2 = acc1*dinv + b1, two features per lane per k-chunk.
  v2f af[4];
#pragma unroll
  for (int k = 0; k < 4; ++k) {
    const int kb = 4 * k + koff;
    v2f xv = *(const v2f*)(xrow + kb);
    af[k].x = xv.x * di + b1[kb + 0];
    af[k].y = xv.y * di + b1[kb + 1];
  }

  v8f c = {};
#pragma unroll
  for (int k = 0; k < 4; ++k) {
    c = __builtin_amdgcn_wmma_f32_16x16x4_f32(
        /*neg_a=*/false, af[k], /*neg_b=*/false, bf[k],
        /*c_mod=*/(short)0, c, /*reuse_a=*/false, /*reuse_b=*/false);
  }

  // Epilogue scale: dinv[nb+m] lives in lane m's `di` -> wave shuffle instead
  // of extra global loads. All lanes active here (shuffle-safe).
  const int mbase = (lane >> 4) * 8;
  float vout[8];
#pragma unroll
  for (int r = 0; r < 8; ++r)
    vout[r] = c[r] * __shfl(di, mbase + r, 32);

  if (row < 2) {                           // only output cols 0,1 are real
#pragma unroll
    for (int r = 0; r < 8; ++r) {
      const int m = nb + mbase + r;
      hs2[(size_t)m * 2 + row]  = vout[r];
      acc2[(size_t)m * 2 + row] = vout[r]; // self-loop contribution
    }
  }
}

// --- layer-2 edge scatter: acc2[dst] += hs2[src] (2 feats) -----------------
__global__ void gcn_scatter_l2_kernel(const int* __restrict__ src,
                                      const int* __restrict__ dst,
                                      const float* __restrict__ hs2,
                                      float* __restrict__ acc2, int e) {
  int i = blockIdx.x * blockDim.x + threadIdx.x;
  if (i >= e) return;
  int s = src[i], d = dst[i];
  float2 m = ((const float2*)hs2)[s];
  atomicAdd(&acc2[(size_t)d * 2 + 0], m.x);
  atomicAdd(&acc2[(size_t)d * 2 + 1], m.y);
}

// --- global mean pool: sums[g] += acc2[i]*dinv[i] + b2 ; cnt[g] += 1 -------
__global__ void gcn_pool_kernel(const int* __restrict__ batch,
                                const float* __restrict__ acc2,
                                const float* __restrict__ dinv,
                                const float* __restrict__ b2,
                                float* __restrict__ sums,
                                float* __restrict__ cnt, int n) {
  int i = blockIdx.x * blockDim.x + threadIdx.x;
  if (i >= n) return;
  int g = batch[i];
  float di = dinv[i];
  float v0 = acc2[(size_t)i * 2 + 0] * di + b2[0];
  float v1 = acc2[(size_t)i * 2 + 1] * di + b2[1];
  atomicAdd(&sums[g * 2 + 0], v0);
  atomicAdd(&sums[g * 2 + 1], v1);
  atomicAdd(&cnt[g], 1.0f);
}

__global__ void gcn_final_kernel(const float* __restrict__ sums,
                                 const float* __restrict__ cnt,
                                 float* __restrict__ out) {
  int g = threadIdx.x;
  if (g < kGraphs) {
    float c = fmaxf(cnt[g], 1.0f);
    out[g * 2 + 0] = sums[g * 2 + 0] / c;
    out[g * 2 + 1] = sums[g * 2 + 1] / c;
  }
}

extern "C" void kernel_launch(void* const* d_in, const int* in_sizes, int n_in,
                              void* d_out, int out_size, void* d_ws, size_t ws_size,
                              hipStream_t stream) {
  const float* x   = (const float*)d_in[0];
  const int*   ei  = (const int*)d_in[1];   // edge_index [2, E] flat
  const int*   bat = (const int*)d_in[2];
  const float* W1  = (const float*)d_in[3];
  const float* b1  = (const float*)d_in[4];
  const float* W2  = (const float*)d_in[5];
  const float* b2  = (const float*)d_in[6];
  float* out = (float*)d_out;

  const int N = kNodes, E = kEdges;
  const int* src = ei;
  const int* dst = ei + E;

  // workspace carve-out (~15.3 MB total), 256B-aligned slices
  char* ws = (char*)d_ws;
  size_t off = 0;
  auto carve = [&](size_t bytes) -> void* {
    void* p = ws + off;
    off += (bytes + 255) & ~(size_t)255;
    return p;
  };
  unsigned* degu = (unsigned*)carve((size_t)N * 4);
  float* dinv = (float*)carve((size_t)N * 4);
  float* hs1  = (float*)carve((size_t)N * 64);
  float* acc1 = (float*)carve((size_t)N * 64);
  float* hs2  = (float*)carve((size_t)N * 8);
  float* acc2 = (float*)carve((size_t)N * 8);
  float* sums = (float*)carve((size_t)2 * kGraphs * 4);
  float* cnt  = (float*)carve((size_t)kGraphs * 4);

  dim3 blk(256);
  int gN = (N + 255) / 256;
  int gE = (E + 255) / 256;

  gcn_init_kernel<<<gN, blk, 0, stream>>>(degu, sums, cnt, N);
  gcn_deg_kernel<<<gE, blk, 0, stream>>>(dst, degu, E);
  gcn_node_l1_kernel<<<gN, blk, 0, stream>>>(x, degu, W1, dinv, hs1, acc1, N);
  gcn_scatter_l1_kernel<<<gE, blk, 0, stream>>>(src, dst, hs1, acc1, E);

  int nTiles = (N + 15) / 16;              // 6250, exact
  int gT = (nTiles + 7) / 8;               // 8 waves (tiles) per 256-thread block
  gcn_l2_wmma_kernel<<<gT, blk, 0, stream>>>(acc1, dinv, b1, W2, hs2, acc2, nTiles);

  gcn_scatter_l2_kernel<<<gE, blk, 0, stream>>>(src, dst, hs2, acc2, E);
  gcn_pool_kernel<<<gN, blk, 0, stream>>>(bat, acc2, dinv, b2, sums, cnt, N);
  gcn_final_kernel<<<1, 64, 0, stream>>>(sums, cnt, out);
}